// LuongAttnDecoderRNN_79517024518150
// MI455X (gfx1250) — compile-verified
//
#include <hip/hip_runtime.h>
#include <math.h>

// Problem constants
#define Bq    64
#define Lq    2048
#define Hq    512
#define Vq    50000
#define FourH 2048

typedef float v2f __attribute__((ext_vector_type(2)));
typedef float v4f __attribute__((ext_vector_type(4)));
typedef float v8f __attribute__((ext_vector_type(8)));

// ---------------------------------------------------------------------------
// 1) Embedding gather: emb[b,h] = embedding[seq[b], h]
// ---------------------------------------------------------------------------
__global__ void k_embed(const int* __restrict__ seq,
                        const float* __restrict__ table,
                        float* __restrict__ emb) {
    int i = blockIdx.x * blockDim.x + threadIdx.x;   // 0..B*H-1
    int b = i >> 9;                                  // /512
    int h = i & (Hq - 1);
    emb[i] = table[(size_t)seq[b] * Hq + h];
}

// ---------------------------------------------------------------------------
// 2) LSTM gates GEMM: gates = emb @ W_ih^T + h0 @ W_hh^T   (bias added here)
//    M=64, N=2048, K=512.  One wave per 16x16 tile, fp32 WMMA (exact).
// ---------------------------------------------------------------------------
__global__ void k_gates(const float* __restrict__ emb,
                        const float* __restrict__ h0,
                        const float* __restrict__ Wih,
                        const float* __restrict__ Whh,
                        const float* __restrict__ bih,
                        const float* __restrict__ bhh,
                        float* __restrict__ gates) {
    const int lane = threadIdx.x;
    const int row  = lane & 15;
    const int half = lane >> 4;
    const int n0 = blockIdx.x * 16;   // output-feature tile (0..2047)
    const int m0 = blockIdx.y * 16;   // batch tile (0..63)

    const float* aP = emb + (size_t)(m0 + row) * Hq + 2 * half;
    const float* hP = h0  + (size_t)(m0 + row) * Hq + 2 * half;
    const float* bP = Wih + (size_t)(n0 + row) * Hq + 2 * half;
    const float* cP = Whh + (size_t)(n0 + row) * Hq + 2 * half;

    v8f acc = {};
    for (int k = 0; k < Hq; k += 4) {
        v2f a  = *(const v2f*)(aP + k);
        v2f bt = *(const v2f*)(bP + k);
        acc = __builtin_amdgcn_wmma_f32_16x16x4_f32(false, a, false, bt,
                                                    (short)0, acc, false, false);
        v2f a2  = *(const v2f*)(hP + k);
        v2f bt2 = *(const v2f*)(cP + k);
        acc = __builtin_amdgcn_wmma_f32_16x16x4_f32(false, a2, false, bt2,
                                                    (short)0, acc, false, false);
    }

    const int col = lane & 15;
    const float bias = bih[n0 + col] + bhh[n0 + col];
#pragma unroll
    for (int r = 0; r < 8; ++r) {
        int m = m0 + r + 8 * half;
        gates[(size_t)m * FourH + n0 + col] = acc[r] + bias;
    }
}

// ---------------------------------------------------------------------------
// 3) LSTM cell elementwise: i,f,g,o -> c_new, h_new
// ---------------------------------------------------------------------------
__global__ void k_cell(const float* __restrict__ gates,
                       const float* __restrict__ c0,
                       float* __restrict__ h_ws,
                       float* __restrict__ h_out,
                       float* __restrict__ c_out) {
    int i = blockIdx.x * blockDim.x + threadIdx.x;   // 0..B*H-1
    int b = i >> 9;
    int h = i & (Hq - 1);
    const float* g = gates + (size_t)b * FourH;
    float gi = g[h];
    float gf = g[Hq + h];
    float gg = g[2 * Hq + h];
    float go = g[3 * Hq + h];
    float si = 1.f / (1.f + expf(-gi));
    float sf = 1.f / (1.f + expf(-gf));
    float so = 1.f / (1.f + expf(-go));
    float tg = tanhf(gg);
    float cn = sf * c0[i] + si * tg;
    float hn = so * tanhf(cn);
    c_out[i] = cn;
    h_out[i] = hn;
    h_ws[i]  = hn;
}

// ---------------------------------------------------------------------------
// 4) Fused attention: ONE pass over enc[L,B,H] (256 MB).
//    softmax is over the batch axis -> local to each l-slice.
//    grid = NWG blocks x 1024 threads; each block streams L/NWG l-slices.
//    Energy-phase read is RT (caches the 128 KB slice); ctx-phase re-read of
//    the same slice is nontemporal (last use). Context partials live in 8
//    v4f registers/thread, spilled once to ws.
// ---------------------------------------------------------------------------
__global__ void __launch_bounds__(1024)
k_attn(const float* __restrict__ enc,
       const float* __restrict__ h_new,
       float* __restrict__ attn_out,   // [B, L]
       float* __restrict__ ctx_part,   // [NWG][B*H]
       int lPerBlk) {
    __shared__ float s_part[1024];
    __shared__ float s_e[Bq];
    __shared__ float s_w[Bq];

    const int t = threadIdx.x;
    const int b = t >> 4;          // energy batch index (0..63)
    const int p = t & 15;          // 32-float h-segment

    v4f acc4[8];
#pragma unroll
    for (int j = 0; j < 8; ++j) acc4[j] = (v4f){0.f, 0.f, 0.f, 0.f};

    const v4f* hv = (const v4f*)(h_new + (size_t)b * Hq + p * 32);
    v4f hreg[8];
#pragma unroll
    for (int q = 0; q < 8; ++q) hreg[q] = hv[q];   // h_new slice stays in regs

    const int l0 = blockIdx.x * lPerBlk;
    for (int li = 0; li < lPerBlk; ++li) {
        const int l = l0 + li;
        const float* encl = enc + (size_t)l * (Bq * Hq);

        // --- energies: e[b] = sum_h h_new[b,h]*enc[l,b,h] (RT: warm cache)
        const v4f* ev = (const v4f*)(encl + (size_t)b * Hq + p * 32);
        float e = 0.f;
#pragma unroll
        for (int q = 0; q < 8; ++q) {
            v4f x = ev[q];
            v4f y = hreg[q];
            e += x.x * y.x + x.y * y.y + x.z * y.z + x.w * y.w;
        }
        s_part[t] = e;
        __syncthreads();

        if (t < Bq) {
            float s = 0.f;
#pragma unroll
            for (int q = 0; q < 16; ++q) s += s_part[t * 16 + q];
            s_e[t] = s;
        }
        __syncthreads();

        // --- softmax across batch (64 values) ---
        if (t < Bq) {
            float mx = -3.0e38f;
            for (int bb = 0; bb < Bq; ++bb) mx = fmaxf(mx, s_e[bb]);
            float den = 0.f;
            for (int bb = 0; bb < Bq; ++bb) den += expf(s_e[bb] - mx);
            float w = expf(s_e[t] - mx) / den;
            s_w[t] = w;
            __builtin_nontemporal_store(w, &attn_out[(size_t)t * Lq + l]);
        }
        __syncthreads();

        // --- context accumulation: thread owns v4f elements t + 1024*j ---
        const v4f* e4 = (const v4f*)encl;
#pragma unroll
        for (int j = 0; j < 8; ++j) {
            int i4 = t + (j << 10);              // v4f index, 0..8191
            float w = s_w[i4 >> 7];              // 128 v4f per batch row
            v4f x = __builtin_nontemporal_load(&e4[i4]);   // last use of slice
            acc4[j] += w * x;
        }
        __syncthreads();   // s_part reused next iteration
    }

    v4f* op = (v4f*)(ctx_part + (size_t)blockIdx.x * (Bq * Hq));
#pragma unroll
    for (int j = 0; j < 8; ++j) op[t + (j << 10)] = acc4[j];
}

// ---------------------------------------------------------------------------
// 5) Reduce context partials (NWG blocks worth) -> context
// ---------------------------------------------------------------------------
__global__ void k_ctxred(const float* __restrict__ part,
                         float* __restrict__ ctx_ws,
                         float* __restrict__ ctx_out,
                         int nwg) {
    int i = blockIdx.x * blockDim.x + threadIdx.x;   // v4f index 0..8191
    const v4f* p4 = (const v4f*)part;
    v4f s = (v4f){0.f, 0.f, 0.f, 0.f};
    for (int w = 0; w < nwg; ++w) s += p4[(size_t)w * (Bq * Hq / 4) + i];
    ((v4f*)ctx_ws)[i]  = s;
    ((v4f*)ctx_out)[i] = s;
}

// ---------------------------------------------------------------------------
// 6) concat projection: tanh([h_new|ctx] @ W_concat^T + b_concat)
//    M=64, N=512, K=1024 (K split: first 512 from h_new, last 512 from ctx)
// ---------------------------------------------------------------------------
__global__ void k_concat(const float* __restrict__ h_new,
                         const float* __restrict__ ctx,
                         const float* __restrict__ Wc,
                         const float* __restrict__ bc,
                         float* __restrict__ cat_out) {
    const int lane = threadIdx.x;
    const int row  = lane & 15;
    const int half = lane >> 4;
    const int n0 = blockIdx.x * 16;
    const int m0 = blockIdx.y * 16;

    const float* bRow = Wc + (size_t)(n0 + row) * (2 * Hq) + 2 * half;
    const float* aH = h_new + (size_t)(m0 + row) * Hq + 2 * half;
    const float* aC = ctx   + (size_t)(m0 + row) * Hq + 2 * half;

    v8f acc = {};
    for (int k = 0; k < Hq; k += 4) {
        v2f a  = *(const v2f*)(aH + k);
        v2f bt = *(const v2f*)(bRow + k);
        acc = __builtin_amdgcn_wmma_f32_16x16x4_f32(false, a, false, bt,
                                                    (short)0, acc, false, false);
    }
    for (int k = 0; k < Hq; k += 4) {
        v2f a  = *(const v2f*)(aC + k);
        v2f bt = *(const v2f*)(bRow + Hq + k);
        acc = __builtin_amdgcn_wmma_f32_16x16x4_f32(false, a, false, bt,
                                                    (short)0, acc, false, false);
    }

    const int col = lane & 15;
    const float bb = bc[n0 + col];
#pragma unroll
    for (int r = 0; r < 8; ++r) {
        int m = m0 + r + 8 * half;
        cat_out[(size_t)m * Hq + n0 + col] = tanhf(acc[r] + bb);
    }
}

// ---------------------------------------------------------------------------
// 7) Output projection: out = cat @ W_out^T + b_out
//    M=64, N=50000 (=3125*16), K=512. One wave per N-tile; the loaded W_out
//    B-tile is reused across all 4 M-tiles -> W_out streamed exactly once
//    (nontemporal: keep the 192 MB L2 for everything else).
// ---------------------------------------------------------------------------
__global__ void k_out(const float* __restrict__ cat,
                      const float* __restrict__ Wo,
                      const float* __restrict__ bo,
                      float* __restrict__ out) {
    const int lane = threadIdx.x;
    const int row  = lane & 15;
    const int half = lane >> 4;
    const int n0 = blockIdx.x * 16;

    const float* bP = Wo  + (size_t)(n0 + row) * Hq + 2 * half;
    const float* a0 = cat + (size_t)(0 * 16 + row) * Hq + 2 * half;
    const float* a1 = cat + (size_t)(1 * 16 + row) * Hq + 2 * half;
    const float* a2 = cat + (size_t)(2 * 16 + row) * Hq + 2 * half;
    const float* a3 = cat + (size_t)(3 * 16 + row) * Hq + 2 * half;

    v8f c0 = {}, c1 = {}, c2 = {}, c3 = {};
    for (int k = 0; k < Hq; k += 4) {
        v2f bt = __builtin_nontemporal_load((const v2f*)(bP + k));
        v2f x0 = *(const v2f*)(a0 + k);
        c0 = __builtin_amdgcn_wmma_f32_16x16x4_f32(false, x0, false, bt,
                                                   (short)0, c0, false, false);
        v2f x1 = *(const v2f*)(a1 + k);
        c1 = __builtin_amdgcn_wmma_f32_16x16x4_f32(false, x1, false, bt,
                                                   (short)0, c1, false, false);
        v2f x2 = *(const v2f*)(a2 + k);
        c2 = __builtin_amdgcn_wmma_f32_16x16x4_f32(false, x2, false, bt,
                                                   (short)0, c2, false, false);
        v2f x3 = *(const v2f*)(a3 + k);
        c3 = __builtin_amdgcn_wmma_f32_16x16x4_f32(false, x3, false, bt,
                                                   (short)0, c3, false, false);
    }

    const int col = lane & 15;
    const float bias = bo[n0 + col];
#pragma unroll
    for (int r = 0; r < 8; ++r) {
        int mr = r + 8 * half;
        __builtin_nontemporal_store(c0[r] + bias,
                                    &out[(size_t)(0 * 16 + mr) * Vq + n0 + col]);
        __builtin_nontemporal_store(c1[r] + bias,
                                    &out[(size_t)(1 * 16 + mr) * Vq + n0 + col]);
        __builtin_nontemporal_store(c2[r] + bias,
                                    &out[(size_t)(2 * 16 + mr) * Vq + n0 + col]);
        __builtin_nontemporal_store(c3[r] + bias,
                                    &out[(size_t)(3 * 16 + mr) * Vq + n0 + col]);
    }
}

// ---------------------------------------------------------------------------
// Launch
// ---------------------------------------------------------------------------
extern "C" void kernel_launch(void* const* d_in, const int* in_sizes, int n_in,
                              void* d_out, int out_size, void* d_ws, size_t ws_size,
                              hipStream_t stream) {
    const int*   seq   = (const int*)  d_in[0];
    // d_in[1] = last_context: unused by the reference computation
    const float* h0    = (const float*)d_in[2];
    const float* c0    = (const float*)d_in[3];
    const float* enc   = (const float*)d_in[4];
    const float* table = (const float*)d_in[5];
    const float* Wih   = (const float*)d_in[6];
    const float* Whh   = (const float*)d_in[7];
    const float* bih   = (const float*)d_in[8];
    const float* bhh   = (const float*)d_in[9];
    const float* Wc    = (const float*)d_in[10];
    const float* bc    = (const float*)d_in[11];
    const float* Wo    = (const float*)d_in[12];
    const float* bo    = (const float*)d_in[13];

    float* out = (float*)d_out;
    // Output tuple layout (flat, return order):
    float* o_output = out;                              // [B,V]
    float* o_ctx    = out + (size_t)Bq * Vq;            // [B,H]
    float* o_h      = o_ctx + (size_t)Bq * Hq;          // [1,B,H]
    float* o_c      = o_h   + (size_t)Bq * Hq;          // [1,B,H]
    float* o_attn   = o_c   + (size_t)Bq * Hq;          // [B,1,L]

    // Workspace layout (floats)
    float* ws       = (float*)d_ws;
    float* w_emb    = ws;                                // 32768
    float* w_gates  = w_emb   + (size_t)Bq * Hq;         // 131072
    float* w_h      = w_gates + (size_t)Bq * FourH;      // 32768
    float* w_ctx    = w_h     + (size_t)Bq * Hq;         // 32768
    float* w_cat    = w_ctx   + (size_t)Bq * Hq;         // 32768
    float* w_part   = w_cat   + (size_t)Bq * Hq;         // NWG * 32768
    const size_t fixedF = (size_t)(w_part - ws);         // 262144 floats

    // Size the attention grid from the available scratch (deterministic:
    // ws_size is constant across calls). NWG = power of two <= 128.
    int nwg = 1;
    size_t availF = ws_size / sizeof(float);
    if (availF > fixedF) {
        size_t cap = (availF - fixedF) / (size_t)(Bq * Hq);
        while ((size_t)(nwg * 2) <= cap && nwg * 2 <= 128) nwg *= 2;
    }
    const int lPerBlk = Lq / nwg;   // powers of two divide 2048 exactly

    // 1) embedding gather
    k_embed<<<(Bq * Hq) / 256, 256, 0, stream>>>(seq, table, w_emb);
    // 2) gates GEMM (fp32 WMMA)
    k_gates<<<dim3(FourH / 16, Bq / 16), 32, 0, stream>>>(
        w_emb, h0, Wih, Whh, bih, bhh, w_gates);
    // 3) LSTM cell
    k_cell<<<(Bq * Hq) / 256, 256, 0, stream>>>(w_gates, c0, w_h, o_h, o_c);
    // 4) fused attention, single pass over enc
    k_attn<<<nwg, 1024, 0, stream>>>(enc, w_h, o_attn, w_part, lPerBlk);
    // 5) context partial reduction
    k_ctxred<<<(Bq * Hq / 4) / 256, 256, 0, stream>>>(w_part, w_ctx, o_ctx, nwg);
    // 6) concat projection (fp32 WMMA)
    k_concat<<<dim3(Hq / 16, Bq / 16), 32, 0, stream>>>(
        w_h, w_ctx, Wc, bc, w_cat);
    // 7) output projection (fp32 WMMA, W_out streamed once, NT)
    k_out<<<Vq / 16, 32, 0, stream>>>(w_cat, Wo, bo, o_output);
}